// LNCCLoss_15539191677017
// MI455X (gfx1250) — compile-verified
//
#include <hip/hip_runtime.h>

#define BB 2
#define DD 160
#define HH 192
#define WW 160
#define NVOX (BB*DD*HH*WW)          /* 9,830,400 */
#define RC   (1.0f/729.0f)

typedef __attribute__((ext_vector_type(2))) float v2f;
typedef __attribute__((ext_vector_type(8))) float v8f;

// ---------------------------------------------------------------------------
// Async global->LDS helper. CDNA5: LDS flat addr low 32 bits == workgroup-
// relative LDS offset, so truncating the generic pointer gives the VDST value.
// ---------------------------------------------------------------------------
__device__ __forceinline__ void async_ld(const float* g, float* l) {
  unsigned lds_addr = (unsigned)(size_t)l;
  unsigned long long ga = (unsigned long long)(size_t)g;
  asm volatile("global_load_async_to_lds_b32 %0, %1, off"
               :: "v"(lds_addr), "v"(ga) : "memory");
}
__device__ __forceinline__ void wait_async() {
  asm volatile("s_wait_asynccnt 0x0" ::: "memory");
}

// ---------------------------------------------------------------------------
// Band matrix for 9-tap box as 16x16 <- Band(16x24) x In(24x16), K split into
// six 16x16x4 f32 WMMAs.  A layout (32-bit A 16x4): lanes 0-15 M=lane,
// VGPR j <-> K=j ; lanes 16-31 M=lane-16, VGPR j <-> K=j+2.
// ---------------------------------------------------------------------------
__device__ __forceinline__ void make_band(v2f* a, int m, int hi) {
#pragma unroll
  for (int kk = 0; kk < 6; ++kk) {
    int r0 = kk*4 + 2*hi;          // K index of vector element 0 for this lane
    int r1 = r0 + 1;
    v2f t;
    t.x = (r0 >= m && r0 <= m + 8) ? 1.0f : 0.0f;
    t.y = (r1 >= m && r1 <= m + 8) ? 1.0f : 0.0f;
    a[kk] = t;
  }
}

// 16 outputs along a strided axis x 16 columns, via 6 chained WMMAs.
// B layout assumed symmetric to A: lane L, VGPR j holds B[K=j+2*(L>=16)][N=L&15].
__device__ __forceinline__ v8f box16(const float* __restrict__ in, int base,
                                     int ax0, int L, int axStride, int col,
                                     int hi, const v2f* a) {
  v8f c = {0.f,0.f,0.f,0.f,0.f,0.f,0.f,0.f};
#pragma unroll
  for (int kk = 0; kk < 6; ++kk) {
    int r0 = ax0 - 4 + kk*4 + 2*hi;
    int r1 = r0 + 1;
    int ra = r0 < 0 ? 0 : (r0 > L-1 ? L-1 : r0);   // replicate padding
    int rb = r1 < 0 ? 0 : (r1 > L-1 ? L-1 : r1);
    v2f b;
    b.x = in[base + ra*axStride + col];
    b.y = in[base + rb*axStride + col];
    c = __builtin_amdgcn_wmma_f32_16x16x4_f32(false, a[kk], false, b,
                                              (short)0, c, false, false);
  }
  return c;
}

// ---------------------------------------------------------------------------
// P1: W-axis box sum of t and p (blockIdx.y selects field). Wave-per-line,
// line staged in LDS via async loads, edge-replicated in LDS.
// ---------------------------------------------------------------------------
__global__ __launch_bounds__(256) void lncc_wpass(const float* __restrict__ t,
                                                  const float* __restrict__ p,
                                                  float* __restrict__ o0,
                                                  float* __restrict__ o1) {
  __shared__ float smem[8 * 168];
  const int wave = threadIdx.x >> 5, lane = threadIdx.x & 31;
  const int line = blockIdx.x * 8 + wave;            // < 61440
  const float* in  = blockIdx.y ? p  : t;
  float*       out = blockIdx.y ? o1 : o0;
  const size_t lb = (size_t)line * WW;
  const int wb = wave * 168;
#pragma unroll
  for (int k = 0; k < 5; ++k) {
    int idx = lane + 32*k;
    async_ld(in + lb + idx, &smem[wb + 4 + idx]);
  }
  wait_async();
  if (lane == 0) { float v = smem[wb+4];   smem[wb+0]=v; smem[wb+1]=v; smem[wb+2]=v; smem[wb+3]=v; }
  if (lane == 1) { float v = smem[wb+163]; smem[wb+164]=v; smem[wb+165]=v; smem[wb+166]=v; smem[wb+167]=v; }
  __syncthreads();
#pragma unroll
  for (int k = 0; k < 5; ++k) {
    int i = lane + 32*k;
    float s = 0.f;
#pragma unroll
    for (int j = 0; j < 9; ++j) s += smem[wb + i + j];
    out[lb + i] = s;
  }
}

// ---------------------------------------------------------------------------
// P4: pointwise centering + products + W-axis box sum, fused.
// Reads t,p and the full box sums S_t,S_p; writes W-box of u1=tc*pc,u2=tc^2,u3=pc^2.
// ---------------------------------------------------------------------------
__global__ __launch_bounds__(256) void lncc_wpass_u(
    const float* __restrict__ t,  const float* __restrict__ p,
    const float* __restrict__ st, const float* __restrict__ sp,
    float* __restrict__ o1, float* __restrict__ o2, float* __restrict__ o3) {
  __shared__ float smem[8 * 7 * 168];
  const int wave = threadIdx.x >> 5, lane = threadIdx.x & 31;
  const int line = blockIdx.x * 8 + wave;
  const size_t lb = (size_t)line * WW;
  const int wb = wave * 7 * 168;
  const float* ins[4] = { t, p, st, sp };
#pragma unroll
  for (int f = 0; f < 4; ++f) {
    const float* lp = ins[f] + lb;
#pragma unroll
    for (int k = 0; k < 5; ++k) {
      int idx = lane + 32*k;
      async_ld(lp + idx, &smem[wb + f*168 + 4 + idx]);
    }
  }
  wait_async();
  if (lane < 4) {                       // left edge replicate, 4 input lines
    int L0 = wb + lane*168;
    float v = smem[L0+4];
    smem[L0+0]=v; smem[L0+1]=v; smem[L0+2]=v; smem[L0+3]=v;
  } else if (lane < 8) {                // right edge replicate
    int L0 = wb + (lane-4)*168;
    float v = smem[L0+163];
    smem[L0+164]=v; smem[L0+165]=v; smem[L0+166]=v; smem[L0+167]=v;
  }
  __syncthreads();
  for (int idx = lane; idx < 168; idx += 32) {
    float tc = smem[wb +        idx] - smem[wb + 2*168 + idx] * RC;
    float pc = smem[wb + 168  + idx] - smem[wb + 3*168 + idx] * RC;
    smem[wb + 4*168 + idx] = tc * pc;
    smem[wb + 5*168 + idx] = tc * tc;
    smem[wb + 6*168 + idx] = pc * pc;
  }
  __syncthreads();
#pragma unroll
  for (int k = 0; k < 5; ++k) {
    int i = lane + 32*k;
    float s1 = 0.f, s2 = 0.f, s3 = 0.f;
#pragma unroll
    for (int j = 0; j < 9; ++j) {
      s1 += smem[wb + 4*168 + i + j];
      s2 += smem[wb + 5*168 + i + j];
      s3 += smem[wb + 6*168 + i + j];
    }
    o1[lb + i] = s1; o2[lb + i] = s2; o3[lb + i] = s3;
  }
}

// ---------------------------------------------------------------------------
// P2/P3/P5: generic strided-axis 9-tap box pass as WMMA stencil, up to 3
// fields per wave-tile.  base = (o/OA)*M1 + (o%OA)*M2.
// ---------------------------------------------------------------------------
__global__ __launch_bounds__(256) void lncc_spass(
    const float* __restrict__ i0, const float* __restrict__ i1, const float* __restrict__ i2,
    float* __restrict__ o0, float* __restrict__ o1, float* __restrict__ o2,
    int nf, int L, int axStride, int outerCount, int OA, int M1, int M2,
    int tilesPerOuter) {
  const int wave = threadIdx.x >> 5, lane = threadIdx.x & 31;
  const int tid = blockIdx.x * 8 + wave;
  if (tid >= outerCount * tilesPerOuter) return;
  const int o   = tid / tilesPerOuter;
  const int rem = tid % tilesPerOuter;
  const int axT = rem / 10, wT = rem % 10;
  const int base = (o / OA) * M1 + (o % OA) * M2;
  const int ax0 = axT * 16, w0 = wT * 16;
  const int m = lane & 15, hi = lane >> 4;
  const int col = w0 + m;
  v2f a[6];
  make_band(a, m, hi);
  const float* ins[3]  = { i0, i1, i2 };
  float*       outs[3] = { o0, o1, o2 };
#pragma unroll
  for (int f = 0; f < 3; ++f) {
    if (f < nf) {
      v8f c = box16(ins[f], base, ax0, L, axStride, col, hi, a);
#pragma unroll
      for (int r = 0; r < 8; ++r)
        outs[f][(size_t)base + (size_t)(ax0 + r + 8*hi) * axStride + col] = c[r];
    }
  }
}

// ---------------------------------------------------------------------------
// P6: D-axis WMMA box pass on the three product fields, fused with cc and a
// block reduction (one partial per block).  D geometry hardcoded.
// ---------------------------------------------------------------------------
__global__ __launch_bounds__(256) void lncc_final(
    const float* __restrict__ fA, const float* __restrict__ fB,
    const float* __restrict__ fC, float* __restrict__ partials) {
  __shared__ float red[256];
  const int wave = threadIdx.x >> 5, lane = threadIdx.x & 31;
  const int tid = blockIdx.x * 8 + wave;             // < 38400 (grid exact)
  const int o = tid / 100, rem = tid % 100;
  const int axT = rem / 10, wT = rem % 10;
  const int base = (o / HH) * (DD*HH*WW) + (o % HH) * WW;
  const int ax0 = axT * 16, w0 = wT * 16;
  const int m = lane & 15, hi = lane >> 4;
  const int col = w0 + m;
  v2f a[6];
  make_band(a, m, hi);
  v8f cA = box16(fA, base, ax0, DD, HH*WW, col, hi, a);
  v8f cB = box16(fB, base, ax0, DD, HH*WW, col, hi, a);
  v8f cC = box16(fC, base, ax0, DD, HH*WW, col, hi, a);
  float s = 0.f;
#pragma unroll
  for (int r = 0; r < 8; ++r) {
    float n = cA[r];
    float v = (n*n + 1e-6f) / (cB[r]*cC[r] + 1e-6f);
    v = fminf(fmaxf(v, 0.0f), 1.0f);
    s += v;
  }
  red[threadIdx.x] = s;
  __syncthreads();
  for (int stp = 128; stp > 0; stp >>= 1) {
    if ((int)threadIdx.x < stp) red[threadIdx.x] += red[threadIdx.x + stp];
    __syncthreads();
  }
  if (threadIdx.x == 0) partials[blockIdx.x] = red[0];
}

// P7: final reduction -> -mean(cc)
__global__ __launch_bounds__(256) void lncc_reduce(const float* __restrict__ partials,
                                                   int n, float* __restrict__ out) {
  __shared__ double red[256];
  double s = 0.0;
  for (int i = threadIdx.x; i < n; i += 256) s += (double)partials[i];
  red[threadIdx.x] = s;
  __syncthreads();
  for (int stp = 128; stp > 0; stp >>= 1) {
    if ((int)threadIdx.x < stp) red[threadIdx.x] += red[threadIdx.x + stp];
    __syncthreads();
  }
  if (threadIdx.x == 0) out[0] = (float)(-red[0] / (double)NVOX);
}

// ---------------------------------------------------------------------------
extern "C" void kernel_launch(void* const* d_in, const int* in_sizes, int n_in,
                              void* d_out, int out_size, void* d_ws, size_t ws_size,
                              hipStream_t stream) {
  (void)in_sizes; (void)n_in; (void)out_size; (void)ws_size;
  const float* t = (const float*)d_in[0];
  const float* p = (const float*)d_in[1];
  float* out = (float*)d_out;
  float* ws = (float*)d_ws;
  const size_t FS = (size_t)NVOX;
  float* b0 = ws;          float* b1 = ws + FS;     float* b2 = ws + 2*FS;
  float* b3 = ws + 3*FS;   float* b4 = ws + 4*FS;   float* b5 = ws + 5*FS;
  float* partials = ws + 6*FS;                      // 4800 floats

  // P1: W box of t,p -> b0,b1
  lncc_wpass<<<dim3(7680, 2), 256, 0, stream>>>(t, p, b0, b1);
  // P2: H box (L=192, stride=W) -> b2,b3   [outer o = b*D+d : base = o*H*W]
  lncc_spass<<<4800, 256, 0, stream>>>(b0, b1, nullptr, b2, b3, nullptr,
                                       2, HH, WW, BB*DD, BB*DD, 0, HH*WW,
                                       (HH/16)*10);
  // P3: D box (L=160, stride=H*W) -> b0,b1 = S_t,S_p  [o = b*H+h]
  lncc_spass<<<4800, 256, 0, stream>>>(b2, b3, nullptr, b0, b1, nullptr,
                                       2, DD, HH*WW, BB*HH, HH, DD*HH*WW, WW,
                                       (DD/16)*10);
  // P4: center + products + W box -> b2,b3,b4
  lncc_wpass_u<<<7680, 256, 0, stream>>>(t, p, b0, b1, b2, b3, b4);
  // P5: H box of products -> b0,b1,b5
  lncc_spass<<<4800, 256, 0, stream>>>(b2, b3, b4, b0, b1, b5,
                                       3, HH, WW, BB*DD, BB*DD, 0, HH*WW,
                                       (HH/16)*10);
  // P6: D box + cc + per-block partial sums
  lncc_final<<<4800, 256, 0, stream>>>(b0, b1, b5, partials);
  // P7: -mean
  lncc_reduce<<<1, 256, 0, stream>>>(partials, 4800, out);
}